// WMMSE_block_90391881712214
// MI455X (gfx1250) — compile-verified
//
#include <hip/hip_runtime.h>

// MI455X / gfx1250 WMMSE unfolding step.
// One workgroup (256 threads = 8 wave32) per batch element b.
// fp32 WMMA (v_wmma_f32_16x16x4_f32) for the batched Phi matmuls:
//   V = Phi @ U   ( [64,64] @ [64,32] complex, all 16 k batched in N )
//   Q = Phi^H @ P ( same shape )
// Everything else (skinny N=2 contractions, 4x4 Hermitian-PD solves) on VALU.

typedef float v2f __attribute__((ext_vector_type(2)));
typedef float v8f __attribute__((ext_vector_type(8)));

struct cx { float re, im; };
__device__ __forceinline__ cx cmul(cx a, cx b)  { return cx{a.re*b.re - a.im*b.im, a.re*b.im + a.im*b.re}; }
__device__ __forceinline__ cx cmulc(cx a, cx b) { return cx{a.re*b.re + a.im*b.im, a.im*b.re - a.re*b.im}; } // a*conj(b)
__device__ __forceinline__ cx cadd(cx a, cx b)  { return cx{a.re + b.re, a.im + b.im}; }
__device__ __forceinline__ cx csub(cx a, cx b)  { return cx{a.re - b.re, a.im - b.im}; }
__device__ __forceinline__ cx cinvf(cx a) { float d = 1.0f / (a.re*a.re + a.im*a.im); return cx{a.re*d, -a.im*d}; }

// Gauss-Jordan solve of A X = B for 4x4 complex Hermitian-PD A, 2 RHS.
// (HPD -> nonzero leading minors, unpivoted elimination is valid.)
__device__ void gj_solve4x2(cx A[4][4], cx B[4][2]) {
  for (int c = 0; c < 4; ++c) {
    cx inv = cinvf(A[c][c]);
    for (int j = c; j < 4; ++j) A[c][j] = cmul(A[c][j], inv);
    B[c][0] = cmul(B[c][0], inv);
    B[c][1] = cmul(B[c][1], inv);
    for (int r = 0; r < 4; ++r) {
      if (r == c) continue;
      cx f = A[r][c];
      for (int j = c; j < 4; ++j) A[r][j] = csub(A[r][j], cmul(f, A[c][j]));
      B[r][0] = csub(B[r][0], cmul(f, B[c][0]));
      B[r][1] = csub(B[r][1], cmul(f, B[c][1]));
    }
  }
}

// Complex [64,64] x [64,32] matmul via fp32 WMMA 16x16x4.
// CONJT=false: O = Phi @ B ;  CONJT=true: O = Phi^H @ B (conj-transpose).
// Phi read from global (row-major [64][64]); B and O in LDS:
// layout [row*32 + col], imag part at +2048 floats.
// 8 waves -> 8 output 16x16 complex tiles (4 M-tiles x 2 N-tiles).
template <bool CONJT>
__device__ __forceinline__ void phi_matmul(const float* __restrict__ phiRe,
                                           const float* __restrict__ phiIm,
                                           const float* __restrict__ Bbuf,
                                           float* __restrict__ Obuf, int tid) {
  const int wid  = tid >> 5;        // 0..7
  const int lane = tid & 31;
  const int mi = wid & 3, ni = wid >> 2;
  const int lm = lane & 15, lh = lane >> 4;
  const int arow = mi * 16 + lm;    // A fragment row (M)
  const int bcol = ni * 16 + lm;    // B fragment col (N)

  v8f s1 = {0,0,0,0,0,0,0,0};       // sum Are*Bre
  v8f s2 = {0,0,0,0,0,0,0,0};       // sum Aim*Bim
  v8f s3 = {0,0,0,0,0,0,0,0};       // sum Are*Bim
  v8f s4 = {0,0,0,0,0,0,0,0};       // sum Aim*Bre
#pragma unroll
  for (int j = 0; j < 16; ++j) {    // K = 64 in steps of 4
    const int c0 = j * 4 + lh * 2;  // this lane's K pair
    v2f are, aim, bre, bim;
    if (!CONJT) {
      are[0] = phiRe[arow * 64 + c0];       are[1] = phiRe[arow * 64 + c0 + 1];
      aim[0] = phiIm[arow * 64 + c0];       aim[1] = phiIm[arow * 64 + c0 + 1];
    } else {  // A[m][k] = Phi[k][m] (conj handled in combine signs)
      are[0] = phiRe[c0 * 64 + arow];       are[1] = phiRe[(c0 + 1) * 64 + arow];
      aim[0] = phiIm[c0 * 64 + arow];       aim[1] = phiIm[(c0 + 1) * 64 + arow];
    }
    bre[0] = Bbuf[c0 * 32 + bcol];          bre[1] = Bbuf[(c0 + 1) * 32 + bcol];
    bim[0] = Bbuf[2048 + c0 * 32 + bcol];   bim[1] = Bbuf[2048 + (c0 + 1) * 32 + bcol];
    s1 = __builtin_amdgcn_wmma_f32_16x16x4_f32(false, are, false, bre, (short)0, s1, false, false);
    s2 = __builtin_amdgcn_wmma_f32_16x16x4_f32(false, aim, false, bim, (short)0, s2, false, false);
    s3 = __builtin_amdgcn_wmma_f32_16x16x4_f32(false, are, false, bim, (short)0, s3, false, false);
    s4 = __builtin_amdgcn_wmma_f32_16x16x4_f32(false, aim, false, bre, (short)0, s4, false, false);
  }
#pragma unroll
  for (int v = 0; v < 8; ++v) {
    const int row = mi * 16 + v + 8 * lh;
    if (!CONJT) {  // (re + i im) = (s1 - s2) + i (s3 + s4)
      Obuf[row * 32 + bcol]        = s1[v] - s2[v];
      Obuf[2048 + row * 32 + bcol] = s3[v] + s4[v];
    } else {       // conj(Phi^T): (s1 + s2) + i (s3 - s4)
      Obuf[row * 32 + bcol]        = s1[v] + s2[v];
      Obuf[2048 + row * 32 + bcol] = s3[v] - s4[v];
    }
  }
}

__global__ __launch_bounds__(256) void wmmse_step_kernel(
    const float* __restrict__ H1re, const float* __restrict__ H1im,   // [B,64,32,16]
    const float* __restrict__ H2re, const float* __restrict__ H2im,   // [B,32,64,16]
    const float* __restrict__ PhiRe, const float* __restrict__ PhiIm, // [B,64,64]
    const float* __restrict__ FrfRe, const float* __restrict__ FrfIm, // [B,32,4]
    const float* __restrict__ WrfRe, const float* __restrict__ WrfIm, // [B,32,4]
    const float* __restrict__ FbbRe, const float* __restrict__ FbbIm, // [16,B,4,2]
    const float* __restrict__ Xi,                                     // [16,B]
    float* __restrict__ Out)                                          // [16,B,4,2,2]
{
  // Reusable 16KB LDS buffers ("im part" always at +2048 floats)
  __shared__ float smem0[4096];  // U -> G -> Z -> Gt
  __shared__ float smem1[4096];  // V -> Q
  __shared__ float smem2[4096];  // Y -> P
  __shared__ float sFrfRe[128], sFrfIm[128], sWrfRe[128], sWrfIm[128];
  __shared__ float sWhGre[128], sWhGim[128], sWbbRe[128], sWbbIm[128];
  __shared__ float sLamRe[64],  sLamIm[64],  sBetaRe[16], sBetaIm[16];
  __shared__ float sSre[128],   sSim[128];

  const int tid = threadIdx.x;
  const int b   = blockIdx.x;
  const float* phiReB = PhiRe + (size_t)b * 4096;
  const float* phiImB = PhiIm + (size_t)b * 4096;

  // ---- Phase 0: stage Frf, Wrf ([32][4] re/im each) ----
  if (tid < 128) {
    sFrfRe[tid] = FrfRe[b * 128 + tid];
    sFrfIm[tid] = FrfIm[b * 128 + tid];
    sWrfRe[tid] = WrfRe[b * 128 + tid];
    sWrfIm[tid] = WrfIm[b * 128 + tid];
  }
  __syncthreads();

  // ---- A0: Y[k][t][s] = (Frf @ Fbb_k)[t][s] / xi_k   -> smem2[t*32+k*2+s] ----
  for (int it = 0; it < 2; ++it) {
    const int idx = it * 256 + tid;     // 512 units
    const int k = idx & 15, t = idx >> 4;   // t<32
    const float inv_xi = 1.0f / Xi[k * 256 + b];
    for (int s = 0; s < 2; ++s) {
      float ar = 0.f, ai = 0.f;
      for (int r = 0; r < 4; ++r) {
        const float fr = sFrfRe[t * 4 + r], fi = sFrfIm[t * 4 + r];
        const size_t g = (((size_t)k * 256 + b) * 4 + r) * 2 + s;
        const float br = FbbRe[g], bi = FbbIm[g];
        ar += fr * br - fi * bi;
        ai += fr * bi + fi * br;
      }
      smem2[t * 32 + k * 2 + s]        = ar * inv_xi;
      smem2[2048 + t * 32 + k * 2 + s] = ai * inv_xi;
    }
  }
  __syncthreads();

  // ---- A1: U[p][2k+s] = sum_t h1[p][t][k] * Y[k][t][s]  -> smem0 ----
  for (int it = 0; it < 4; ++it) {
    const int idx = it * 256 + tid;     // 1024 units
    const int k = idx & 15, p = idx >> 4;   // p<64
    float ur0 = 0, ui0 = 0, ur1 = 0, ui1 = 0;
    for (int t = 0; t < 32; ++t) {
      const size_t g = (((size_t)b * 64 + p) * 32 + t) * 16 + k;
      const float hr = H1re[g], hi = H1im[g];
      const float y0r = smem2[t * 32 + k * 2],        y0i = smem2[2048 + t * 32 + k * 2];
      const float y1r = smem2[t * 32 + k * 2 + 1],    y1i = smem2[2048 + t * 32 + k * 2 + 1];
      ur0 += hr * y0r - hi * y0i;  ui0 += hr * y0i + hi * y0r;
      ur1 += hr * y1r - hi * y1i;  ui1 += hr * y1i + hi * y1r;
    }
    smem0[p * 32 + k * 2]            = ur0;  smem0[2048 + p * 32 + k * 2]     = ui0;
    smem0[p * 32 + k * 2 + 1]        = ur1;  smem0[2048 + p * 32 + k * 2 + 1] = ui1;
  }
  __syncthreads();

  // ---- B (WMMA): V = Phi @ U   (smem0 -> smem1) ----
  phi_matmul<false>(phiReB, phiImB, smem0, smem1, tid);
  __syncthreads();

  // ---- C1: G[k][r][s] = sum_q h2[r][q][k] * V[q][2k+s]  -> smem0 ----
  for (int it = 0; it < 2; ++it) {
    const int idx = it * 256 + tid;
    const int k = idx & 15, r = idx >> 4;   // r<32
    float gr0 = 0, gi0 = 0, gr1 = 0, gi1 = 0;
    for (int q = 0; q < 64; ++q) {
      const size_t g = (((size_t)b * 32 + r) * 64 + q) * 16 + k;
      const float hr = H2re[g], hi = H2im[g];
      const float v0r = smem1[q * 32 + k * 2],     v0i = smem1[2048 + q * 32 + k * 2];
      const float v1r = smem1[q * 32 + k * 2 + 1], v1i = smem1[2048 + q * 32 + k * 2 + 1];
      gr0 += hr * v0r - hi * v0i;  gi0 += hr * v0i + hi * v0r;
      gr1 += hr * v1r - hi * v1i;  gi1 += hr * v1i + hi * v1r;
    }
    smem0[r * 32 + k * 2]            = gr0;  smem0[2048 + r * 32 + k * 2]     = gi0;
    smem0[r * 32 + k * 2 + 1]        = gr1;  smem0[2048 + r * 32 + k * 2 + 1] = gi1;
  }
  __syncthreads();

  // ---- C2: WhG[k][i][s] = sum_r conj(Wrf[r][i]) * G[k][r][s] ----
  if (tid < 128) {
    const int k = tid >> 3, i = (tid >> 1) & 3, s = tid & 1;
    float wr = 0, wi = 0;
    for (int r = 0; r < 32; ++r) {
      const float ar = sWrfRe[r * 4 + i], ai = sWrfIm[r * 4 + i];
      const float gr = smem0[r * 32 + k * 2 + s], gi = smem0[2048 + r * 32 + k * 2 + s];
      wr += ar * gr + ai * gi;
      wi += ar * gi - ai * gr;
    }
    sWhGre[k * 8 + i * 2 + s] = wr;
    sWhGim[k * 8 + i * 2 + s] = wi;
  }
  __syncthreads();

  // ---- C3 (per k): A = WhG WhG^H + aI ; Wbb = A^-1 WhG ; Lam ; Beta ----
  if (tid < 16) {
    const int k = tid;
    const float xiv = Xi[k * 256 + b];
    const float alpha = 0.8f / (xiv * xiv);   // N_POWER*Nr/Nrf_r = 0.1*32/4
    cx W[4][2];
    for (int i = 0; i < 4; ++i)
      for (int s = 0; s < 2; ++s)
        W[i][s] = cx{sWhGre[k * 8 + i * 2 + s], sWhGim[k * 8 + i * 2 + s]};
    cx A[4][4];
    for (int i = 0; i < 4; ++i)
      for (int j = 0; j < 4; ++j) {
        cx acc = cadd(cmulc(W[i][0], W[j][0]), cmulc(W[i][1], W[j][1]));
        if (i == j) acc.re += alpha;
        A[i][j] = acc;
      }
    cx Bm[4][2];
    for (int i = 0; i < 4; ++i) { Bm[i][0] = W[i][0]; Bm[i][1] = W[i][1]; }
    gj_solve4x2(A, Bm);                       // Bm = Wbb
    // Lam[s][u] = d(s,u) + sum_i conj(W[i][s]) W[i][u] / alpha
    const float ia = 1.0f / alpha;
    cx L[2][2];
    for (int s = 0; s < 2; ++s)
      for (int u = 0; u < 2; ++u) {
        cx acc{0.f, 0.f};
        for (int i = 0; i < 4; ++i) acc = cadd(acc, cmulc(W[i][u], W[i][s]));
        acc.re *= ia; acc.im *= ia;
        if (s == u) acc.re += 1.0f;
        L[s][u] = acc;
      }
    // Beta = 6.4 * trace(Lam @ Wbb^H @ Wbb)
    cx tr{0.f, 0.f};
    for (int s = 0; s < 2; ++s)
      for (int u = 0; u < 2; ++u) {
        cx m{0.f, 0.f};  // M[u][s] = sum_i conj(Wbb[i][u]) Wbb[i][s]
        for (int i = 0; i < 4; ++i) m = cadd(m, cmulc(Bm[i][s], Bm[i][u]));
        tr = cadd(tr, cmul(L[s][u], m));
      }
    sBetaRe[k] = 6.4f * tr.re;                // N_POWER*Nt*Nr/(Nrf_t*Nrf_r)
    sBetaIm[k] = 6.4f * tr.im;
    for (int i = 0; i < 4; ++i)
      for (int s = 0; s < 2; ++s) {
        sWbbRe[k * 8 + i * 2 + s] = Bm[i][s].re;
        sWbbIm[k * 8 + i * 2 + s] = Bm[i][s].im;
      }
    for (int s = 0; s < 2; ++s)
      for (int u = 0; u < 2; ++u) {
        sLamRe[k * 4 + s * 2 + u] = L[s][u].re;
        sLamIm[k * 4 + s * 2 + u] = L[s][u].im;
      }
  }
  __syncthreads();

  // ---- C4: Z[k][r][s] = sum_i Wrf[r][i] * Wbb[k][i][s]  -> smem0 (over G) ----
  for (int it = 0; it < 2; ++it) {
    const int idx = it * 256 + tid;
    const int k = idx & 15, r = idx >> 4;
    for (int s = 0; s < 2; ++s) {
      float zr = 0, zi = 0;
      for (int i = 0; i < 4; ++i) {
        const float ar = sWrfRe[r * 4 + i], ai = sWrfIm[r * 4 + i];
        const float br = sWbbRe[k * 8 + i * 2 + s], bi = sWbbIm[k * 8 + i * 2 + s];
        zr += ar * br - ai * bi;
        zi += ar * bi + ai * br;
      }
      smem0[r * 32 + k * 2 + s]        = zr;
      smem0[2048 + r * 32 + k * 2 + s] = zi;
    }
  }
  __syncthreads();

  // ---- C5: P[q][2k+s] = sum_r conj(h2[r][q][k]) * Z[k][r][s]  -> smem2 ----
  for (int it = 0; it < 4; ++it) {
    const int idx = it * 256 + tid;
    const int k = idx & 15, q = idx >> 4;   // q<64
    float pr0 = 0, pi0 = 0, pr1 = 0, pi1 = 0;
    for (int r = 0; r < 32; ++r) {
      const size_t g = (((size_t)b * 32 + r) * 64 + q) * 16 + k;
      const float hr = H2re[g], hi = H2im[g];
      const float z0r = smem0[r * 32 + k * 2],     z0i = smem0[2048 + r * 32 + k * 2];
      const float z1r = smem0[r * 32 + k * 2 + 1], z1i = smem0[2048 + r * 32 + k * 2 + 1];
      pr0 += hr * z0r + hi * z0i;  pi0 += hr * z0i - hi * z0r;
      pr1 += hr * z1r + hi * z1i;  pi1 += hr * z1i - hi * z1r;
    }
    smem2[q * 32 + k * 2]            = pr0;  smem2[2048 + q * 32 + k * 2]     = pi0;
    smem2[q * 32 + k * 2 + 1]        = pr1;  smem2[2048 + q * 32 + k * 2 + 1] = pi1;
  }
  __syncthreads();

  // ---- D (WMMA): Q = Phi^H @ P   (smem2 -> smem1) ----
  phi_matmul<true>(phiReB, phiImB, smem2, smem1, tid);
  __syncthreads();

  // ---- E1: Gt[k][t][s] = sum_p conj(h1[p][t][k]) * Q[p][2k+s] -> smem0 ----
  for (int it = 0; it < 2; ++it) {
    const int idx = it * 256 + tid;
    const int k = idx & 15, t = idx >> 4;   // t<32
    float r0 = 0, i0 = 0, r1 = 0, i1 = 0;
    for (int p = 0; p < 64; ++p) {
      const size_t g = (((size_t)b * 64 + p) * 32 + t) * 16 + k;
      const float hr = H1re[g], hi = H1im[g];
      const float q0r = smem1[p * 32 + k * 2],     q0i = smem1[2048 + p * 32 + k * 2];
      const float q1r = smem1[p * 32 + k * 2 + 1], q1i = smem1[2048 + p * 32 + k * 2 + 1];
      r0 += hr * q0r + hi * q0i;  i0 += hr * q0i - hi * q0r;
      r1 += hr * q1r + hi * q1i;  i1 += hr * q1i - hi * q1r;
    }
    smem0[t * 32 + k * 2]            = r0;  smem0[2048 + t * 32 + k * 2]     = i0;
    smem0[t * 32 + k * 2 + 1]        = r1;  smem0[2048 + t * 32 + k * 2 + 1] = i1;
  }
  __syncthreads();

  // ---- E2: S[k][i][s] = sum_t conj(Frf[t][i]) * Gt[k][t][s] ----
  if (tid < 128) {
    const int k = tid >> 3, i = (tid >> 1) & 3, s = tid & 1;
    float sr = 0, si = 0;
    for (int t = 0; t < 32; ++t) {
      const float fr = sFrfRe[t * 4 + i], fi = sFrfIm[t * 4 + i];
      const float gr = smem0[t * 32 + k * 2 + s], gi = smem0[2048 + t * 32 + k * 2 + s];
      sr += fr * gr + fi * gi;
      si += fr * gi - fi * gr;
    }
    sSre[k * 8 + i * 2 + s] = sr;
    sSim[k * 8 + i * 2 + s] = si;
  }
  __syncthreads();

  // ---- E3 (per k): FhGL = S Lam ; Ftil = FhGL S^H + beta I ; T = Ftil^-1 FhGL ----
  if (tid < 16) {
    const int k = tid;
    cx S[4][2], L[2][2];
    for (int i = 0; i < 4; ++i)
      for (int s = 0; s < 2; ++s)
        S[i][s] = cx{sSre[k * 8 + i * 2 + s], sSim[k * 8 + i * 2 + s]};
    for (int s = 0; s < 2; ++s)
      for (int u = 0; u < 2; ++u)
        L[s][u] = cx{sLamRe[k * 4 + s * 2 + u], sLamIm[k * 4 + s * 2 + u]};
    const cx beta{sBetaRe[k], sBetaIm[k]};
    cx F[4][2];  // FhGL
    for (int i = 0; i < 4; ++i)
      for (int s = 0; s < 2; ++s)
        F[i][s] = cadd(cmul(S[i][0], L[0][s]), cmul(S[i][1], L[1][s]));
    cx A[4][4];
    for (int i = 0; i < 4; ++i)
      for (int j = 0; j < 4; ++j) {
        cx acc = cadd(cmulc(F[i][0], S[j][0]), cmulc(F[i][1], S[j][1]));
        if (i == j) acc = cadd(acc, beta);
        A[i][j] = acc;
      }
    gj_solve4x2(A, F);  // F = T
    float fro2 = 0.f;
    for (int i = 0; i < 4; ++i)
      for (int s = 0; s < 2; ++s)
        fro2 += F[i][s].re * F[i][s].re + F[i][s].im * F[i][s].im;
    const float scale = 0.5f * rsqrtf(fro2);  // (1/sqrt(Nt/Nrf_t/Ns)) / fro
    for (int i = 0; i < 4; ++i)
      for (int s = 0; s < 2; ++s) {
        const size_t o = ((((size_t)k * 256 + b) * 4 + i) * 2 + s) * 2;
        Out[o]     = scale * F[i][s].re;
        Out[o + 1] = scale * F[i][s].im;
      }
  }
}

extern "C" void kernel_launch(void* const* d_in, const int* in_sizes, int n_in,
                              void* d_out, int out_size, void* d_ws, size_t ws_size,
                              hipStream_t stream) {
  const float* H1re  = (const float*)d_in[0];
  const float* H1im  = (const float*)d_in[1];
  const float* H2re  = (const float*)d_in[2];
  const float* H2im  = (const float*)d_in[3];
  const float* PhiRe = (const float*)d_in[4];
  const float* PhiIm = (const float*)d_in[5];
  const float* FrfRe = (const float*)d_in[6];
  const float* FrfIm = (const float*)d_in[7];
  const float* WrfRe = (const float*)d_in[8];
  const float* WrfIm = (const float*)d_in[9];
  const float* FbbRe = (const float*)d_in[10];
  const float* FbbIm = (const float*)d_in[11];
  const float* Xi    = (const float*)d_in[12];
  (void)in_sizes; (void)n_in; (void)out_size; (void)d_ws; (void)ws_size;

  wmmse_step_kernel<<<256, 256, 0, stream>>>(
      H1re, H1im, H2re, H2im, PhiRe, PhiIm,
      FrfRe, FrfIm, WrfRe, WrfIm, FbbRe, FbbIm, Xi, (float*)d_out);
}